// FullFastSTU_48404281426451
// MI455X (gfx1250) — compile-verified
//
#include <hip/hip_runtime.h>
#include <hip/hip_bf16.h>

// ---------------------------------------------------------------------------
// FullFastSTU for MI455X (gfx1250, wave32, WMMA).
//
// Phases (all on `stream`, deterministic, graph-capture safe):
//  K1 stu_scan_localend : per-(n,chunk) local scan endpoints (h from 0)
//  K2 stu_carry         : sequential carry across 16 chunks (h_in per chunk)
//  K3 stu_emit_U        : re-scan chunk from carry, Y = h@C via WMMA f32
//                         16x16x4 (per-d batched matmuls) + FIR, write U
//  K4 stu_gemm          : out = U (4096 x 12288) @ W (12288 x 256) via WMMA
//                         f32 16x16x4, W = concat(M_phi_plus, M_phi_minus)
//
// Roofline: U is 201 MB; write+read ~ 17us at 23.3 TB/s vs ~0.2ms+ of f32
// WMMA compute for the 25.8 GFLOP GEMM -> unfused 2-phase design is optimal.
// Workspace requirement: ~210 MB (U + 2 x 4MB scan buffers).
// ---------------------------------------------------------------------------

#define BSZ   4
#define LSEQ  1024
#define DDIM  256
#define SDIM  64
#define K2T   48      // 2*K
#define KXF   8       // FIR taps
#define NCH   16      // chunks over L
#define LCH   64      // chunk length (NCH*LCH == LSEQ)
#define TLS   16      // sub-chunk (timesteps buffered in LDS per round)
#define KTOT  12288   // GEMM K = K2T * DDIM

typedef float v2f __attribute__((ext_vector_type(2)));
typedef float v8f __attribute__((ext_vector_type(8)));

static __device__ __forceinline__ v8f wmma_f32_16x16x4(v2f a, v2f b, v8f c) {
  // D = A(16x4,f32) * B(4x16,f32) + C(16x16,f32); VOP3P v_wmma_f32_16x16x4_f32
  return __builtin_amdgcn_wmma_f32_16x16x4_f32(false, a, false, b,
                                               (short)0, c, false, false);
}

// ---------------------------------------------------------------------------
// K1: local chunk endpoints. grid(16 chunks, 32 n-groups), 256 threads.
// thread = (dl = t&31 -> sequence within group, sg = t>>5 -> 8 states each).
// ---------------------------------------------------------------------------
__global__ void stu_scan_localend(const float* __restrict__ x,
                                  const float* __restrict__ A,
                                  const float* __restrict__ Bv,
                                  float* __restrict__ le) {
  __shared__ float xs1[LCH * 32];
  const int t = threadIdx.x;
  const int chunk = blockIdx.x;
  const int ngrp  = blockIdx.y;

  for (int i = t; i < LCH * 32; i += 256) {
    const int step = i >> 5, dl = i & 31;
    const int n = ngrp * 32 + dl;
    const int b = n >> 8, d = n & 255;
    const int l = chunk * LCH + step;
    xs1[i] = x[((size_t)(b * LSEQ + l)) * DDIM + d];
  }
  __syncthreads();

  const int dl = t & 31, sg = t >> 5;
  const int n = ngrp * 32 + dl;
  float areg[8], breg[8], h[8];
#pragma unroll
  for (int j = 0; j < 8; ++j) {
    const int s = sg * 8 + j;
    areg[j] = A[s]; breg[j] = Bv[s]; h[j] = 0.0f;
  }
  for (int step = 0; step < LCH; ++step) {
    const float xv = xs1[step * 32 + dl];
#pragma unroll
    for (int j = 0; j < 8; ++j) h[j] = fmaf(h[j], areg[j], xv * breg[j]);
  }
  const size_t base = ((size_t)n * NCH + chunk) * SDIM + sg * 8;
#pragma unroll
  for (int j = 0; j < 8; ++j) le[base + j] = h[j];
}

// ---------------------------------------------------------------------------
// K2: carry pass. 65536 threads: thread = (n = g/64, s = g&63).
// carry[n][c][s] = h entering chunk c;  carry[c] = A^64 * carry[c-1] + le[c-1]
// ---------------------------------------------------------------------------
__global__ void stu_carry(const float* __restrict__ A,
                          const float* __restrict__ le,
                          float* __restrict__ carry) {
  const int g = blockIdx.x * 256 + threadIdx.x;
  const int n = g >> 6, s = g & 63;
  float p = A[s];
#pragma unroll
  for (int i = 0; i < 6; ++i) p = p * p;   // A[s]^64
  const size_t base = (size_t)n * NCH * SDIM + s;
  float c = 0.0f;
  carry[base] = 0.0f;
  for (int ch = 1; ch < NCH; ++ch) {
    c = fmaf(p, c, le[base + (size_t)(ch - 1) * SDIM]);
    carry[base + (size_t)ch * SDIM] = c;
  }
}

// ---------------------------------------------------------------------------
// K3: re-scan chunk + Y = h@C via WMMA + FIR, emit U.
// grid(16 chunks, 8 d-groups, 4 b), 256 threads, dynamic LDS:
//   xs2 [32 d][71]           (x with 7-tap halo, stride 71 -> conflict-free)
//   hbuf[32 d][16 l][s:66]   (d-stride 1057 odd -> conflict-free)
//   Cld [64 s][48 k], M2ld[48 k][8 j]
// Per sub-chunk: scan 16 steps into hbuf, then each wave runs 4 d-batched
// (16l x 64s)@(64s x 48k) matmuls as 3 n-tiles x 16 k-steps of WMMA 16x16x4.
// ---------------------------------------------------------------------------
#define XST 71
#define HLS 66
#define HDS 1057
__global__ void stu_emit_U(const float* __restrict__ x,
                           const float* __restrict__ A,
                           const float* __restrict__ Bv,
                           const float* __restrict__ Cm,
                           const float* __restrict__ M,
                           const float* __restrict__ carry,
                           float* __restrict__ U) {
  extern __shared__ float smem[];
  float* xs2  = smem;                    // 32*71   = 2272
  float* hbuf = xs2 + 32 * XST;          // 32*1057 = 33824
  float* Cld  = hbuf + 32 * HDS;         // 64*48   = 3072
  float* M2ld = Cld + SDIM * K2T;        // 48*8    = 384

  const int t = threadIdx.x;
  const int chunk = blockIdx.x, dgrp = blockIdx.y, b = blockIdx.z;
  const int d0 = dgrp * 32, l0 = chunk * LCH;

  for (int i = t; i < XST * 32; i += 256) {
    const int lr = (i >> 5) - 7, dl = i & 31;
    const int l = l0 + lr;
    float v = 0.0f;
    if (l >= 0) v = x[((size_t)(b * LSEQ + l)) * DDIM + d0 + dl];
    xs2[dl * XST + (lr + 7)] = v;
  }
  for (int i = t; i < SDIM * K2T; i += 256) Cld[i] = Cm[i];
  for (int i = t; i < K2T * KXF; i += 256) M2ld[i] = M[i];

  const int dl = t & 31, sg = t >> 5;
  const int n = b * DDIM + d0 + dl;
  float areg[8], breg[8], h[8];
#pragma unroll
  for (int j = 0; j < 8; ++j) {
    const int s = sg * 8 + j;
    areg[j] = A[s]; breg[j] = Bv[s];
  }
  {
    const size_t cb = ((size_t)n * NCH + chunk) * SDIM + sg * 8;
#pragma unroll
    for (int j = 0; j < 8; ++j) h[j] = carry[cb + j];
  }
  __syncthreads();

  const int wv = t >> 5, lane = t & 31;
  const int mlane = lane & 15, hi = lane >> 4;

  for (int sub = 0; sub < LCH / TLS; ++sub) {
    const int bl = sub * TLS;
    // --- scan TLS steps into hbuf ---
    for (int st = 0; st < TLS; ++st) {
      const float xv = xs2[dl * XST + bl + st + 7];
#pragma unroll
      for (int j = 0; j < 8; ++j) {
        h[j] = fmaf(h[j], areg[j], xv * breg[j]);
        hbuf[dl * HDS + st * HLS + sg * 8 + j] = h[j];
      }
    }
    __syncthreads();
    // --- Y phase: wave wv owns d = wv*4 .. wv*4+3 ---
    for (int dd = 0; dd < 4; ++dd) {
      const int d = wv * 4 + dd;
      const float* hb = hbuf + d * HDS;
      for (int nt = 0; nt < 3; ++nt) {
        v8f acc = {0.f, 0.f, 0.f, 0.f, 0.f, 0.f, 0.f, 0.f};
#pragma unroll
        for (int ks = 0; ks < 16; ++ks) {
          const int s0 = ks * 4 + 2 * hi;
          v2f af, bf;
          af[0] = hb[mlane * HLS + s0];
          af[1] = hb[mlane * HLS + s0 + 1];
          bf[0] = Cld[(s0) * K2T + nt * 16 + mlane];
          bf[1] = Cld[(s0 + 1) * K2T + nt * 16 + mlane];
          acc = wmma_f32_16x16x4(af, bf, acc);
        }
        const int kcol = nt * 16 + mlane;
        const float* m2 = M2ld + kcol * KXF;
#pragma unroll
        for (int r = 0; r < 8; ++r) {
          const int lr = bl + r + 8 * hi;   // l within chunk
          float f = 0.0f;
#pragma unroll
          for (int j = 0; j < KXF; ++j)
            f = fmaf(xs2[d * XST + lr + 7 - j], m2[j], f);
          const size_t row = (size_t)(b * LSEQ + l0 + lr);
          U[(row * K2T + kcol) * DDIM + d0 + d] = acc[r] + f;
        }
      }
    }
    __syncthreads();
  }
}

// ---------------------------------------------------------------------------
// K4: out(4096x256) = U(4096x12288) @ W(12288x256), W = [phiP; phiM] rows.
// grid(64 Mtiles, 4 Ntiles), 256 threads. 64x64 tile, K staged 32-deep.
// A staged row-major stride 36 (16B aligned, conflict-free frag reads),
// B staged transposed [n][k] stride 36 so B-fragments are contiguous pairs.
// ---------------------------------------------------------------------------
#define AST 36
__global__ void stu_gemm(const float* __restrict__ U,
                         const float* __restrict__ phiP,
                         const float* __restrict__ phiM,
                         float* __restrict__ out) {
  __shared__ float Al[64 * AST];
  __shared__ float Bt[64 * AST];
  const int t = threadIdx.x;
  const int row0 = blockIdx.x * 64;
  const int col0 = blockIdx.y * 64;
  const int wv = t >> 5, lane = t & 31;
  const int mlane = lane & 15, hi = lane >> 4;
  const int mt = wv & 3, ntb = (wv >> 2) * 2;

  v8f c0 = {0.f, 0.f, 0.f, 0.f, 0.f, 0.f, 0.f, 0.f};
  v8f c1 = {0.f, 0.f, 0.f, 0.f, 0.f, 0.f, 0.f, 0.f};

  for (int kb = 0; kb < KTOT; kb += 32) {
    // stage A tile: 64 rows x 32 k, float4 per thread x2
#pragma unroll
    for (int i = 0; i < 2; ++i) {
      const int f = t + i * 256;          // 0..511
      const int m = f >> 3, kq = (f & 7) * 4;
      const float4 av =
          *(const float4*)(U + (size_t)(row0 + m) * KTOT + kb + kq);
      *(float4*)(Al + m * AST + kq) = av;
    }
    // stage B tile transposed: 32 k x 64 n  ->  Bt[n][k]
#pragma unroll
    for (int i = 0; i < 2; ++i) {
      const int f = t + i * 256;
      const int kr = f >> 4, nq = (f & 15) * 4;
      const int r = kb + kr;
      const float* wrow = (r < 6144) ? (phiP + (size_t)r * 256)
                                     : (phiM + (size_t)(r - 6144) * 256);
      const float4 w4 = *(const float4*)(wrow + col0 + nq);
      Bt[(nq + 0) * AST + kr] = w4.x;
      Bt[(nq + 1) * AST + kr] = w4.y;
      Bt[(nq + 2) * AST + kr] = w4.z;
      Bt[(nq + 3) * AST + kr] = w4.w;
    }
    __syncthreads();
#pragma unroll
    for (int kk = 0; kk < 32; kk += 4) {
      const int s0 = kk + 2 * hi;
      v2f af, b0, b1;
      af[0] = Al[(mt * 16 + mlane) * AST + s0];
      af[1] = Al[(mt * 16 + mlane) * AST + s0 + 1];
      b0[0] = Bt[(ntb * 16 + mlane) * AST + s0];
      b0[1] = Bt[(ntb * 16 + mlane) * AST + s0 + 1];
      b1[0] = Bt[((ntb + 1) * 16 + mlane) * AST + s0];
      b1[1] = Bt[((ntb + 1) * 16 + mlane) * AST + s0 + 1];
      c0 = wmma_f32_16x16x4(af, b0, c0);
      c1 = wmma_f32_16x16x4(af, b1, c1);
    }
    __syncthreads();
  }

#pragma unroll
  for (int r = 0; r < 8; ++r) {
    const int row = row0 + mt * 16 + r + 8 * hi;
    out[(size_t)row * 256 + col0 + ntb * 16 + mlane] = c0[r];
    out[(size_t)row * 256 + col0 + (ntb + 1) * 16 + mlane] = c1[r];
  }
}

// ---------------------------------------------------------------------------
extern "C" void kernel_launch(void* const* d_in, const int* in_sizes, int n_in,
                              void* d_out, int out_size, void* d_ws,
                              size_t ws_size, hipStream_t stream) {
  const float* x    = (const float*)d_in[0];
  const float* A    = (const float*)d_in[1];
  const float* Bv   = (const float*)d_in[2];
  const float* Cm   = (const float*)d_in[3];
  const float* M    = (const float*)d_in[4];
  const float* phiP = (const float*)d_in[5];
  const float* phiM = (const float*)d_in[6];
  float* out = (float*)d_out;

  // workspace layout (~210 MB total)
  float* U     = (float*)d_ws;                           // 4096*12288
  float* le    = U + (size_t)4096 * KTOT;                // 1024*16*64
  float* carry = le + (size_t)1024 * NCH * SDIM;         // 1024*16*64

  stu_scan_localend<<<dim3(NCH, 32), 256, 0, stream>>>(x, A, Bv, le);
  stu_carry<<<dim3(256), 256, 0, stream>>>(A, le, carry);
  const size_t smem3 =
      (size_t)(32 * XST + 32 * HDS + SDIM * K2T + K2T * KXF) * sizeof(float);
  stu_emit_U<<<dim3(NCH, 8, 4), 256, smem3, stream>>>(x, A, Bv, Cm, M, carry, U);
  stu_gemm<<<dim3(64, 4), 256, 0, stream>>>(U, phiP, phiM, out);
}